// QOC_19207093748546
// MI455X (gfx1250) — compile-verified
//
#include <hip/hip_runtime.h>

// ---------------------------------------------------------------------------
// Quantum optimal control propagator for MI455X (gfx1250).
//   U[n] = expm(-i*DT*(Hd + sum_l a[l,n]*MAXPOWER*Hc[l])),  out = prod_n U[n]
// All matmuls are 64x64 complex done with V_WMMA_F32_16X16X4_F32 tiles.
// ---------------------------------------------------------------------------

#define DD 64
#define LDSP 66            // padded LDS row stride (floats) to reduce bank conflicts
#define NSTEPS 8192
#define SEG 64
#define NSEG (NSTEPS / SEG)   // 128 segment products
#define DT_C 1e-4f
#define MP_C 1500.0f
#define TAYLOR_K 12

typedef __attribute__((ext_vector_type(2))) float v2f;
typedef __attribute__((ext_vector_type(8))) float v8f;

// Complex tile matmul-accumulate: (cr,ci) += A(tile-row ti) * B(tile-col tj)
// A,B are complex 64x64 stored row-major (stride LDSP) in LDS.
// Fragment layouts per CDNA5 ISA 7.12.2:
//   A 16x4 : lane&15 = M, K = kb + (lane>>4)*2 + vgpr
//   B 4x16 : lane&15 = N, K = kb + (lane>>4)*2 + vgpr
__device__ __forceinline__ void cmm_tile(
    const float* __restrict__ Ar, const float* __restrict__ Ai,
    const float* __restrict__ Br, const float* __restrict__ Bi,
    int ti, int tj, int lane, v8f& cr, v8f& ci)
{
  const int half = lane >> 4;
  const int mr   = lane & 15;
  const int arow = (ti * 16 + mr) * LDSP;
  const int bcol = tj * 16 + mr;
  for (int ks = 0; ks < 16; ++ks) {
    const int kb = ks * 4 + 2 * half;
    v2f ar, ai, br, bi;
    ar.x = Ar[arow + kb];           ar.y = Ar[arow + kb + 1];
    ai.x = Ai[arow + kb];           ai.y = Ai[arow + kb + 1];
    br.x = Br[kb * LDSP + bcol];    br.y = Br[(kb + 1) * LDSP + bcol];
    bi.x = Bi[kb * LDSP + bcol];    bi.y = Bi[(kb + 1) * LDSP + bcol];
    v2f nai = -ai;  // f32 WMMA has no A-negate modifier; negate fragment in VALU
    cr = __builtin_amdgcn_wmma_f32_16x16x4_f32(false, ar,  false, br, (short)0, cr, false, false);
    cr = __builtin_amdgcn_wmma_f32_16x16x4_f32(false, nai, false, bi, (short)0, cr, false, false);
    ci = __builtin_amdgcn_wmma_f32_16x16x4_f32(false, ar,  false, bi, (short)0, ci, false, false);
    ci = __builtin_amdgcn_wmma_f32_16x16x4_f32(false, ai,  false, br, (short)0, ci, false, false);
  }
}

// Store a C/D fragment (M = v + 8*half, N = lane&15) into LDS tile (ti,tj).
__device__ __forceinline__ void store_tile(float* __restrict__ B, int ti, int tj,
                                           int lane, const v8f& c)
{
  const int half = lane >> 4;
  const int mr   = lane & 15;
#pragma unroll
  for (int v = 0; v < 8; ++v)
    B[(ti * 16 + v + 8 * half) * LDSP + tj * 16 + mr] = c[v];
}

// ---------------------------------------------------------------------------
// Phase 1: each workgroup computes the product of 64 consecutive step unitaries.
// 512 threads = 16 wave32, wave w owns output tile (w>>2, w&3).
// ---------------------------------------------------------------------------
__global__ __launch_bounds__(512) void qoc_segment_kernel(
    const float* __restrict__ a,      // [2, 8192]
    const float* __restrict__ HdRe,   // [64,64]
    const float* __restrict__ HdIm,   // [64,64]
    const float* __restrict__ HcRe,   // [2,64,64]
    const float* __restrict__ HcIm,   // [2,64,64]
    float* __restrict__ ws)           // [NSEG][64*64*2] interleaved (re,im)
{
  __shared__ float Mre[DD * LDSP], Mim[DD * LDSP];  // M, later reused for U
  __shared__ float Pre[DD * LDSP], Pim[DD * LDSP];  // Taylor power term
  __shared__ float Gre[DD * LDSP], Gim[DD * LDSP];  // running segment product

  const int tid  = threadIdx.x;
  const int lane = tid & 31;
  const int w    = tid >> 5;       // wave id 0..15
  const int ti   = w >> 2;
  const int tj   = w & 3;
  const int half = lane >> 4;
  const int mr   = lane & 15;
  const int g    = blockIdx.x;

  for (int t = 0; t < SEG; ++t) {
    const int n = g * SEG + t;
    const float a0 = a[n] * MP_C;
    const float a1 = a[NSTEPS + n] * MP_C;

    // Build M = -i*DT*H:  M_re = DT*Hi, M_im = -DT*Hr
    for (int j = 0; j < 8; ++j) {
      const int e = tid + j * 512;        // 0..4095
      const int r = e >> 6, c = e & 63;
      const float hr = HdRe[e] + a0 * HcRe[e] + a1 * HcRe[4096 + e];
      const float hi = HdIm[e] + a0 * HcIm[e] + a1 * HcIm[4096 + e];
      Mre[r * LDSP + c] = DT_C * hi;
      Mim[r * LDSP + c] = -DT_C * hr;
    }
    __syncthreads();

    // S = I + M (accumulated in registers, C/D fragment layout)
    v8f sr, si;
#pragma unroll
    for (int v = 0; v < 8; ++v) {
      const int row = ti * 16 + v + 8 * half;
      const int col = tj * 16 + mr;
      sr[v] = ((row == col) ? 1.0f : 0.0f) + Mre[row * LDSP + col];
      si[v] = Mim[row * LDSP + col];
    }

    // Taylor: P_k = P_{k-1} * M / k ; S += P_k
    for (int k = 2; k <= TAYLOR_K; ++k) {
      v8f cr = {}; v8f ci = {};
      if (k == 2) cmm_tile(Mre, Mim, Mre, Mim, ti, tj, lane, cr, ci);
      else        cmm_tile(Pre, Pim, Mre, Mim, ti, tj, lane, cr, ci);
      const float invk = 1.0f / (float)k;
#pragma unroll
      for (int v = 0; v < 8; ++v) { cr[v] *= invk; ci[v] *= invk; }
      __syncthreads();                      // all waves done reading old P
      store_tile(Pre, ti, tj, lane, cr);
      store_tile(Pim, ti, tj, lane, ci);
#pragma unroll
      for (int v = 0; v < 8; ++v) { sr[v] += cr[v]; si[v] += ci[v]; }
      __syncthreads();                      // new P visible before next read
    }

    if (t == 0) {
      store_tile(Gre, ti, tj, lane, sr);
      store_tile(Gim, ti, tj, lane, si);
      __syncthreads();
    } else {
      // Publish U into the (now dead) M buffers, then G = G * U
      store_tile(Mre, ti, tj, lane, sr);
      store_tile(Mim, ti, tj, lane, si);
      __syncthreads();
      v8f cr = {}; v8f ci = {};
      cmm_tile(Gre, Gim, Mre, Mim, ti, tj, lane, cr, ci);
      __syncthreads();
      store_tile(Gre, ti, tj, lane, cr);
      store_tile(Gim, ti, tj, lane, ci);
      __syncthreads();
    }
  }

  // Emit segment product (interleaved complex)
  float* out = ws + (size_t)g * (DD * DD * 2);
  for (int j = 0; j < 8; ++j) {
    const int e = tid + j * 512;
    const int r = e >> 6, c = e & 63;
    out[2 * e]     = Gre[r * LDSP + c];
    out[2 * e + 1] = Gim[r * LDSP + c];
  }
}

// ---------------------------------------------------------------------------
// Phase 2: one workgroup folds the 128 segment products left-to-right.
// ---------------------------------------------------------------------------
__global__ __launch_bounds__(512) void qoc_chain_kernel(
    const float* __restrict__ ws, float* __restrict__ out)
{
  __shared__ float Gre[DD * LDSP], Gim[DD * LDSP];
  __shared__ float Ure[DD * LDSP], Uim[DD * LDSP];

  const int tid  = threadIdx.x;
  const int lane = tid & 31;
  const int w    = tid >> 5;
  const int ti   = w >> 2;
  const int tj   = w & 3;

  // G = segment 0
  for (int j = 0; j < 8; ++j) {
    const int e = tid + j * 512;
    const int r = e >> 6, c = e & 63;
    Gre[r * LDSP + c] = ws[2 * e];
    Gim[r * LDSP + c] = ws[2 * e + 1];
  }
  __syncthreads();

  for (int s = 1; s < NSEG; ++s) {
    const float* seg = ws + (size_t)s * (DD * DD * 2);
    for (int j = 0; j < 8; ++j) {
      const int e = tid + j * 512;
      const int r = e >> 6, c = e & 63;
      Ure[r * LDSP + c] = seg[2 * e];
      Uim[r * LDSP + c] = seg[2 * e + 1];
    }
    __syncthreads();
    v8f cr = {}; v8f ci = {};
    cmm_tile(Gre, Gim, Ure, Uim, ti, tj, lane, cr, ci);
    __syncthreads();
    store_tile(Gre, ti, tj, lane, cr);
    store_tile(Gim, ti, tj, lane, ci);
    __syncthreads();
  }

  // Output: complex64 [64,64] -> interleaved float pairs
  for (int j = 0; j < 8; ++j) {
    const int e = tid + j * 512;
    const int r = e >> 6, c = e & 63;
    out[2 * e]     = Gre[r * LDSP + c];
    out[2 * e + 1] = Gim[r * LDSP + c];
  }
}

extern "C" void kernel_launch(void* const* d_in, const int* in_sizes, int n_in,
                              void* d_out, int out_size, void* d_ws, size_t ws_size,
                              hipStream_t stream) {
  const float* a    = (const float*)d_in[0];  // [2,8192]
  const float* HdRe = (const float*)d_in[1];  // [64,64]
  const float* HdIm = (const float*)d_in[2];  // [64,64]
  const float* HcRe = (const float*)d_in[3];  // [2,64,64]
  const float* HcIm = (const float*)d_in[4];  // [2,64,64]
  float* ws  = (float*)d_ws;                  // needs 128*64*64*2*4 = 4 MB
  float* out = (float*)d_out;

  qoc_segment_kernel<<<NSEG, 512, 0, stream>>>(a, HdRe, HdIm, HcRe, HcIm, ws);
  qoc_chain_kernel<<<1, 512, 0, stream>>>(ws, out);
}